// BurgersPINN_1288490188822
// MI455X (gfx1250) — compile-verified
//
#include <hip/hip_runtime.h>
#include <hip/hip_bf16.h>

typedef float v2f __attribute__((ext_vector_type(2)));
typedef float v8f __attribute__((ext_vector_type(8)));

#define H       128
#define AP      132           // padded LDS row stride (words): conflict-free frags
#define TILE    16
#define NLAYER  7
#define NU_F    0.0031830988618367f   // 0.01/pi

__device__ __forceinline__ v8f wmma_f32(v2f a, v2f b, v8f c) {
    // D = A(16x4,f32) * B(4x16,f32) + C(16x16,f32)
    return __builtin_amdgcn_wmma_f32_16x16x4_f32(
        /*neg_a=*/false, a, /*neg_b=*/false, b,
        /*c_mod=*/(short)0, c, /*reuse_a=*/false, /*reuse_b=*/false);
}

// Hardware tanh (v_tanh_f32, 1 TRANS op) with a branchless exp2/rcp fallback.
__device__ __forceinline__ float fast_tanh(float x) {
#if __has_builtin(__builtin_amdgcn_tanhf)
    return __builtin_amdgcn_tanhf(x);
#else
    float ax = __builtin_fabsf(x);
    float t  = __builtin_amdgcn_exp2f(ax * -2.885390081777927f); // -2*log2(e)
    float r  = (1.0f - t) * __builtin_amdgcn_rcpf(1.0f + t);     // tanh(|x|)
    return __builtin_copysignf(r, x);
#endif
}

// Propagate (value, z_x, z_t, z_xx) through the 7 hidden tanh layers.
// act: double-buffered LDS activations [2][4][TILE][AP].
template<bool NG>
__device__ __forceinline__ void hidden_layers(
    float (*act)[4][TILE][AP],
    const float* __restrict__ W_hid,
    const float* __restrict__ b_hid)
{
    const int tid   = threadIdx.x;
    const int lane  = tid & 31;
    const int wv    = tid >> 5;           // wave id 0..7 -> 16-column slice
    const int nl    = lane & 15;
    const int khalf = (lane >> 4) << 1;   // 0 or 2 (A/B fragment K sub-offset)
    const int n     = wv * 16 + nl;       // output column owned by this lane
    const int mbase = (lane >> 4) << 3;   // C fragment row base (0 or 8)

    int cur = 0;
#pragma unroll 1
    for (int l = 0; l < NLAYER; ++l) {
        const float* __restrict__ W = W_hid + l * H * H;
        if (l + 1 < NLAYER) {
            // warm L2/L0 for next layer's weights: 256 threads * 256B = 64KB
            __builtin_prefetch(W_hid + (l + 1) * H * H + tid * 64, 0, 0);
        }

        v8f accv = {}, accx = {}, acct = {}, accxx = {};
#pragma unroll 8
        for (int k0 = 0; k0 < H; k0 += 4) {
            const int kb = k0 + khalf;
            v2f b;                               // B fragment: W[kb..kb+1][n]
            b.x = W[kb * H + n];
            b.y = W[kb * H + H + n];
            v2f a0 = *(const v2f*)&act[cur][0][nl][kb];
            accv = wmma_f32(a0, b, accv);
            if (NG) {
                v2f a1 = *(const v2f*)&act[cur][1][nl][kb];
                v2f a2 = *(const v2f*)&act[cur][2][nl][kb];
                v2f a3 = *(const v2f*)&act[cur][3][nl][kb];
                accx  = wmma_f32(a1, b, accx);
                acct  = wmma_f32(a2, b, acct);
                accxx = wmma_f32(a3, b, accxx);
            }
        }

        const float bias = b_hid[l * H + n];
        const int nxt = cur ^ 1;
#pragma unroll
        for (int v = 0; v < 8; ++v) {
            const int m = mbase + v;             // C layout: row = v + 8*(lane>=16)
            float z = accv[v] + bias;
            float y = fast_tanh(z);
            float g = 1.0f - y * y;
            act[nxt][0][m][n] = y;
            if (NG) {
                float zx  = accx[v];
                float zt  = acct[v];
                float zxx = accxx[v];
                act[nxt][1][m][n] = g * zx;
                act[nxt][2][m][n] = g * zt;
                act[nxt][3][m][n] = g * zxx - 2.0f * y * g * zx * zx;
            }
        }
        __syncthreads();
        cur = nxt;
    }
}

extern "C" __global__ __launch_bounds__(256) void pinn_fused(
    const float* __restrict__ x_f,  const float* __restrict__ x0,
    const float* __restrict__ xbl,  const float* __restrict__ xbr,
    const float* __restrict__ W_in, const float* __restrict__ b_in,
    const float* __restrict__ W_hid,const float* __restrict__ b_hid,
    const float* __restrict__ W_out,const float* __restrict__ b_out,
    float* __restrict__ out,
    int n0r, int nblr, int nbrr)
{
    __shared__ float act[2][4][TILE][AP];
    __shared__ float xin[TILE][2];
    __shared__ float uvals[4][TILE];

    const int tid = threadIdx.x;
    const int T   = blockIdx.x;

    const int t0 = n0r  >> 4;
    const int t1 = t0 + (nblr >> 4);
    const int t2 = t1 + (nbrr >> 4);

    const float* src; int row0; int out_base; bool need_grad;
    if (T < t0)      { src = x0;  row0 = T * TILE;        out_base = T * TILE;                      need_grad = false; }
    else if (T < t1) { src = xbl; row0 = (T - t0) * TILE; out_base = n0r + (T - t0) * TILE;         need_grad = false; }
    else if (T < t2) { src = xbr; row0 = (T - t1) * TILE; out_base = n0r + nblr + (T - t1) * TILE;  need_grad = false; }
    else             { src = x_f; row0 = (T - t2) * TILE; out_base = n0r + nblr + nbrr + (T - t2) * TILE; need_grad = true; }

    if (tid < 2 * TILE) xin[tid >> 1][tid & 1] = src[row0 * 2 + tid];
    __syncthreads();

    // ---- layer 0: 2 -> 128, derivative seeds dz/dx = W_in[0][n], dz/dt = W_in[1][n]
#pragma unroll
    for (int i = 0; i < (TILE * H) / 256; ++i) {   // 2048 (m,n) pairs / 256 threads
        int idx = tid + 256 * i;
        int nn = idx & (H - 1);
        int mm = idx >> 7;
        float w0 = W_in[nn], w1 = W_in[H + nn];
        float z = xin[mm][0] * w0 + xin[mm][1] * w1 + b_in[nn];
        float y = fast_tanh(z);
        float g = 1.0f - y * y;
        act[0][0][mm][nn] = y;
        act[0][1][mm][nn] = g * w0;
        act[0][2][mm][nn] = g * w1;
        act[0][3][mm][nn] = -2.0f * y * g * w0 * w0;
    }
    __syncthreads();

    if (need_grad) hidden_layers<true>(act, W_hid, b_hid);
    else           hidden_layers<false>(act, W_hid, b_hid);

    const int cur = NLAYER & 1;  // final activations buffer
    // ---- final 128 -> 1 per stream
    if (tid < 4 * TILE) {
        int s = tid >> 4, m = tid & 15;
        if (s == 0 || need_grad) {
            float sum = 0.0f;
#pragma unroll 4
            for (int k = 0; k < H; ++k) sum += act[cur][s][m][k] * W_out[k];
            uvals[s][m] = sum;
        }
    }
    __syncthreads();

    if (tid < TILE) {
        int m = tid;
        float u = uvals[0][m] + b_out[0];
        float r;
        if (need_grad) {
            float ux  = uvals[1][m];
            float ut  = uvals[2][m];
            float uxx = uvals[3][m];
            r = ut + u * ux - NU_F * uxx;   // Burgers residual
        } else {
            r = u;
        }
        out[out_base + m] = r;
    }
}

extern "C" void kernel_launch(void* const* d_in, const int* in_sizes, int n_in,
                              void* d_out, int out_size, void* d_ws, size_t ws_size,
                              hipStream_t stream) {
    const float* x_f   = (const float*)d_in[0];
    const float* x0    = (const float*)d_in[1];
    const float* xbl   = (const float*)d_in[2];
    const float* xbr   = (const float*)d_in[3];
    const float* W_in  = (const float*)d_in[4];
    const float* b_in  = (const float*)d_in[5];
    const float* W_hid = (const float*)d_in[6];
    const float* b_hid = (const float*)d_in[7];
    const float* W_out = (const float*)d_in[8];
    const float* b_out = (const float*)d_in[9];
    float* out = (float*)d_out;

    const int nf  = in_sizes[0] / 2;
    const int n0  = in_sizes[1] / 2;
    const int nbl = in_sizes[2] / 2;
    const int nbr = in_sizes[3] / 2;
    const int tiles = (n0 + nbl + nbr + nf) / TILE;   // 8704 for reference sizes

    hipLaunchKernelGGL(pinn_fused, dim3(tiles), dim3(256), 0, stream,
                       x_f, x0, xbl, xbr, W_in, b_in, W_hid, b_hid, W_out, b_out,
                       out, n0, nbl, nbr);
}